// WeightedLCANet_48146583388994
// MI455X (gfx1250) — compile-verified
//
#include <hip/hip_runtime.h>

#define N_BATCH 32768
#define NLEAF   1024
#define LCOLS   1025
#define NODES   2048
#define ROWS_PER_BLOCK 16
#define WAVES_PER_BLOCK 8
#define THREADS (WAVES_PER_BLOCK * 32)
#define XSTRIDE 36   /* padded LDS stride (dwords) per 32-leaf group */

typedef __attribute__((ext_vector_type(2))) float v2f;
typedef __attribute__((ext_vector_type(4))) float v4f;
typedef __attribute__((ext_vector_type(8))) float v8f;

// Non-temporal stores (output is streamed, never re-read: keep L2 clean).
__device__ __forceinline__ void nt_store4(float* p, float a, float b, float c, float d) {
    v4f v = {a, b, c, d};
    __builtin_nontemporal_store(v, (v4f*)p);
}
__device__ __forceinline__ void nt_store2(float* p, float a, float b) {
    v2f v = {a, b};
    __builtin_nontemporal_store(v, (v2f*)p);
}
__device__ __forceinline__ void nt_store1(float* p, float a) {
    __builtin_nontemporal_store(a, p);
}

__launch_bounds__(THREADS)
__global__ void wlca_kernel(const float* __restrict__ X, float* __restrict__ out) {
    constexpr float EPSV = 1e-5f;
    const float MAXV = (float)(1.0 / (1.0 + 1e-5));
    const float B1 = 1.0f - EPSV;          // (1 - eps)
    // powers of (1-eps), constant-folded
    const float cp1 = B1;
    const float cp2 = cp1 * B1;
    const float cp3 = cp2 * B1;
    const float cp4 = cp3 * B1;
    const float cp5 = cp4 * B1;
    const float cp6 = cp5 * B1;
    const float cp7 = cp6 * B1;
    const float cp8 = cp7 * B1;
    const float cp9 = cp8 * B1;
    const float cp10 = cp9 * B1;

    // Per-wave staging region: 32 groups * stride 36 dwords = 1152 floats (4.5KB)
    __shared__ __align__(16) float sX[WAVES_PER_BLOCK * 32 * XSTRIDE];   // 36 KB
    __shared__ __align__(16) float sS[ROWS_PER_BLOCK * 33];              // lane sums, padded

    const int tid = threadIdx.x;
    const int w   = tid >> 5;     // wave id 0..7
    const int l   = tid & 31;     // lane
    float* sx = sX + w * (32 * XSTRIDE);

    // ---------------- Phase A: each wave processes 2 rows ----------------
    #pragma unroll
    for (int rr = 0; rr < 2; ++rr) {
        const int rloc = w * 2 + rr;                       // row within tile, 0..15
        const int r = blockIdx.x * ROWS_PER_BLOCK + rloc;  // global row
        const float* rowX = X + (size_t)r * LCOLS;
        float* orow = out + (size_t)r * NODES;

        // Coalesced NT load of leaf columns 1..1024 into LDS; track running max.
        float mx = -3.402823466e38f;
        #pragma unroll
        for (int j = 0; j < 32; ++j) {
            float v = __builtin_nontemporal_load(&rowX[1 + j * 32 + l]);
            sx[j * XSTRIDE + l] = v;
            mx = fmaxf(mx, v);
        }
        // wave32 max reduction
        #pragma unroll
        for (int off = 16; off >= 1; off >>= 1)
            mx = fmaxf(mx, __shfl_xor(mx, off, 32));

        const float sumL  = mx * (float)NLEAF;
        const float scale = sumL / (sumL + EPSV);

        __builtin_amdgcn_wave_barrier();   // order LDS write->read within wave

        // Contiguous read-back: lane owns leaves [32l, 32l+32)
        float lv[32];
        #pragma unroll
        for (int t = 0; t < 8; ++t) {
            v4f v = *(const v4f*)&sx[l * XSTRIDE + 4 * t];  // 16B aligned LDS b128
            lv[4*t+0] = fmaxf(v[0] - mx + MAXV, 0.0f) * scale;
            lv[4*t+1] = fmaxf(v[1] - mx + MAXV, 0.0f) * scale;
            lv[4*t+2] = fmaxf(v[2] - mx + MAXV, 0.0f) * scale;
            lv[4*t+3] = fmaxf(v[3] - mx + MAXV, 0.0f) * scale;
        }

        // Leaf outputs: nodes 1024..2047 (aligned, non-temporal b128 stores)
        #pragma unroll
        for (int t = 0; t < 8; ++t)
            nt_store4(&orow[1024 + l * 32 + 4 * t],
                      lv[4*t], lv[4*t+1], lv[4*t+2], lv[4*t+3]);

        // Intra-lane tree: depths 9..5 over the lane's 32 contiguous leaves.
        float s2[16];
        #pragma unroll
        for (int p = 0; p < 16; ++p) s2[p] = lv[2*p] + lv[2*p+1];
        #pragma unroll
        for (int t = 0; t < 4; ++t)      // depth 9: nodes 512 + 16l + ...
            nt_store4(&orow[512 + 16 * l + 4 * t],
                      cp1*s2[4*t], cp1*s2[4*t+1], cp1*s2[4*t+2], cp1*s2[4*t+3]);

        float s4[8];
        #pragma unroll
        for (int p = 0; p < 8; ++p) s4[p] = s2[2*p] + s2[2*p+1];
        #pragma unroll
        for (int t = 0; t < 2; ++t)      // depth 8: nodes 256 + 8l + ...
            nt_store4(&orow[256 + 8 * l + 4 * t],
                      cp2*s4[4*t], cp2*s4[4*t+1], cp2*s4[4*t+2], cp2*s4[4*t+3]);

        float s8[4];
        #pragma unroll
        for (int p = 0; p < 4; ++p) s8[p] = s4[2*p] + s4[2*p+1];
        nt_store4(&orow[128 + 4 * l],    // depth 7
                  cp3*s8[0], cp3*s8[1], cp3*s8[2], cp3*s8[3]);

        float s16a = s8[0] + s8[1];
        float s16b = s8[2] + s8[3];
        nt_store2(&orow[64 + 2 * l], cp4*s16a, cp4*s16b);  // depth 6

        float S = s16a + s16b;           // raw lane sum (32 leaves)
        nt_store1(&orow[32 + l], cp5 * S);  // depth 5: node 32 + l

        sS[rloc * 33 + l] = S;           // stash for WMMA cross-lane phase

        if (l == 0)                       // node 0
            nt_store1(&orow[0], EPSV * (fmaxf(rowX[0] + MAXV, 0.0f) + 1.0f));
    }

    __syncthreads();

    // ---------------- Phase B: WMMA f32 16x16x4 cross-lane tree ----------------
    // D[m,n] = sum_k A[m,k] * B[k,n] ; m = batch row in tile, n = node column,
    // k chained over the 32 lane-sums in 8 steps of K=4. (wave 0 only: full EXEC)
    if (tid < 32) {
        const int n    = tid & 15;       // A row (batch) == B column (node) per layout
        const int half = tid >> 4;
        const int koff = half * 2;       // A: V0=K(2*half), V1=K(2*half+1); B same

        // chain1 target node t1 = 1+n (depths 0..3); n==15 is a zero column
        const int t1 = 1 + n;
        const int d  = 31 - __clz(t1);
        const int sh = 5 - d;
        const int tb = t1 - (1 << d);
        float w1 = (d == 0) ? cp10 : (d == 1) ? cp9 : (d == 2) ? cp8 : cp7;
        if (n == 15) w1 = 0.0f;

        v8f acc0 = {};   // nodes 16..31 (depth 4, weight cp6)
        v8f acc1 = {};   // nodes 1..15
        #pragma unroll
        for (int cch = 0; cch < 8; ++cch) {
            const int g0 = 4 * cch + koff;
            const int g1 = g0 + 1;
            v2f a;
            a[0] = sS[n * 33 + g0];
            a[1] = sS[n * 33 + g1];
            v2f b0, b1;
            b0[0] = ((g0 >> 1) == n) ? cp6 : 0.0f;
            b0[1] = ((g1 >> 1) == n) ? cp6 : 0.0f;
            b1[0] = ((g0 >> sh) == tb) ? w1 : 0.0f;
            b1[1] = ((g1 >> sh) == tb) ? w1 : 0.0f;
            acc0 = __builtin_amdgcn_wmma_f32_16x16x4_f32(
                       false, a, false, b0, (short)0, acc0, false, false);
            acc1 = __builtin_amdgcn_wmma_f32_16x16x4_f32(
                       false, a, false, b1, (short)0, acc1, false, false);
        }

        // C/D layout: lane holds column N = tid&15; VGPR v holds M = v + 8*half.
        float* ob = out + (size_t)(blockIdx.x * ROWS_PER_BLOCK) * NODES;
        #pragma unroll
        for (int v = 0; v < 8; ++v) {
            const int mrow = v + 8 * half;
            nt_store1(&ob[(size_t)mrow * NODES + 16 + n], acc0[v]);   // nodes 16..31
            if (n != 15)
                nt_store1(&ob[(size_t)mrow * NODES + 1 + n], acc1[v]); // nodes 1..15
        }
    }
}

extern "C" void kernel_launch(void* const* d_in, const int* in_sizes, int n_in,
                              void* d_out, int out_size, void* d_ws, size_t ws_size,
                              hipStream_t stream) {
    const float* X = (const float*)d_in[0];
    float* out = (float*)d_out;
    (void)in_sizes; (void)n_in; (void)d_ws; (void)ws_size; (void)out_size;
    const int blocks = N_BATCH / ROWS_PER_BLOCK;   // 2048
    wlca_kernel<<<blocks, THREADS, 0, stream>>>(X, out);
}